// DocumentBertScoringLoss_19138374271446
// MI455X (gfx1250) — compile-verified
//
#include <hip/hip_runtime.h>
#include <math.h>

// ---------------------------------------------------------------------------
// Loss = MSE + MR + SIM over B=8192 floats.
// MR term simplifies exactly to mean_{m,n} max(0, 0.1 - |p_m - p_n|):
//   r*dp == |dp| when dp!=0, and r*dp == 0 when dp==0 (hinge value 0.1 either
//   way), so correct_output never enters the pairwise term.
// The 16x16 difference tile DP = p*1^T - 1*p^T is rank-2, so we synthesize it
// with one V_WMMA_F32_16X16X4_F32 per tile:
//   A(16x4) = [p_row | 1 | 0 | 0],  B(4x16) = [1 ; -p_col ; 0 ; 0]
// The per-tile hinge SUM is invariant to the D-matrix lane layout and to any
// consistent K relabeling between A and B, so only the rank-2 construction
// must be exact.
// Roofline: 64 KB of input, ~0.27 GFLOP total -> launch-latency bound; the
// whole predictions vector (32 KB) is staged once per WGP into LDS.
// ---------------------------------------------------------------------------

typedef __attribute__((ext_vector_type(2))) float v2f;
typedef __attribute__((ext_vector_type(8))) float v8f;

#define MR_BIAS   0.1f
#define COS_EPS   1e-8f
#define MR_BLOCKS 64
#define THREADS   256
#define WAVES_PB  (THREADS / 32)   // 8 wave32s per block
#define MAX_N     8192

// ---------------- Kernel 1: O(B) reductions (ssd, dot, |p|^2, |g|^2) --------
__global__ __launch_bounds__(THREADS)
void bert_loss_reduce(const float* __restrict__ p, const float* __restrict__ g,
                      int n, float* __restrict__ ws) {
  int tid = threadIdx.x;
  float ssd = 0.f, dot = 0.f, pp = 0.f, gg = 0.f;
  for (int i = tid; i < n; i += THREADS) {
    float pv = p[i], gv = g[i];
    float df = pv - gv;
    ssd = fmaf(df, df, ssd);
    dot = fmaf(pv, gv, dot);
    pp  = fmaf(pv, pv, pp);
    gg  = fmaf(gv, gv, gg);
  }
  #pragma unroll
  for (int off = 16; off > 0; off >>= 1) {   // wave32 butterfly
    ssd += __shfl_xor(ssd, off, 32);
    dot += __shfl_xor(dot, off, 32);
    pp  += __shfl_xor(pp,  off, 32);
    gg  += __shfl_xor(gg,  off, 32);
  }
  __shared__ float red[WAVES_PB][4];
  int wave = tid >> 5, lane = tid & 31;
  if (lane == 0) {
    red[wave][0] = ssd; red[wave][1] = dot; red[wave][2] = pp; red[wave][3] = gg;
  }
  __syncthreads();
  if (tid == 0) {
    float s0 = 0.f, s1 = 0.f, s2 = 0.f, s3 = 0.f;
    for (int w = 0; w < WAVES_PB; ++w) {
      s0 += red[w][0]; s1 += red[w][1]; s2 += red[w][2]; s3 += red[w][3];
    }
    ws[0] = s0; ws[1] = s1; ws[2] = s2; ws[3] = s3;
  }
}

// ---------------- Kernel 2: pairwise hinge via WMMA rank-2 tiles ------------
__global__ __launch_bounds__(THREADS)
void bert_loss_mr_pairwise(const float* __restrict__ p, int n,
                           float* __restrict__ ws) {
  __shared__ float sp[MAX_N];          // 32 KB: full predictions vector
  __shared__ float red[WAVES_PB];

  int tid = threadIdx.x;
  for (int i = tid; i < n; i += THREADS) sp[i] = p[i];
  __syncthreads();

  const int lane   = tid & 31;
  const int wave   = tid >> 5;
  const int gwave  = blockIdx.x * WAVES_PB + wave;       // wave-uniform
  const int nwaves = MR_BLOCKS * WAVES_PB;               // 512 waves
  const int ntiles = n >> 4;                             // 512 tiles of 16

  const bool  lo    = (lane < 16);
  const int   l16   = lane & 15;
  const float onelo = lo ? 1.0f : 0.0f;  // K=0 row of B / K=1 col of A

  v8f acc = {0.f, 0.f, 0.f, 0.f, 0.f, 0.f, 0.f, 0.f};

  // rt loop bound is wave-uniform -> EXEC stays all-1s for every WMMA.
  for (int rt = gwave; rt < ntiles; rt += nwaves) {
    const float pm = sp[(rt << 4) + l16];
    v2f a;                      // A 16x4: [p | 1 | 0 | 0]
    a.x = lo ? pm : 0.0f;       // lanes 0-15: K=0 ; lanes 16-31: K=2 (zero)
    a.y = onelo;                // lanes 0-15: K=1 ; lanes 16-31: K=3 (zero)

    #pragma unroll 4
    for (int ct = 0; ct < ntiles; ++ct) {
      const float pn = sp[(ct << 4) + l16];
      v2f b;                    // B 4x16: [1 ; -p ; 0 ; 0]
      b.x = onelo;
      b.y = lo ? -pn : 0.0f;

      v8f c = {0.f, 0.f, 0.f, 0.f, 0.f, 0.f, 0.f, 0.f};
      // D = A x B + 0  ->  full 16x16 tile of (p_m - p_n)
      v8f d = __builtin_amdgcn_wmma_f32_16x16x4_f32(
          /*neg_a=*/false, a, /*neg_b=*/false, b,
          /*c_mod=*/(short)0, c, /*reuse_a=*/false, /*reuse_b=*/false);

      #pragma unroll
      for (int k = 0; k < 8; ++k)
        acc[k] += fmaxf(MR_BIAS - fabsf(d[k]), 0.0f);
    }
  }

  float s = 0.f;
  #pragma unroll
  for (int k = 0; k < 8; ++k) s += acc[k];
  #pragma unroll
  for (int off = 16; off > 0; off >>= 1) s += __shfl_xor(s, off, 32);

  if (lane == 0) red[wave] = s;
  __syncthreads();
  if (tid == 0) {
    float t = 0.f;
    for (int w = 0; w < WAVES_PB; ++w) t += red[w];
    ws[4 + blockIdx.x] = t;      // deterministic per-block partial (no atomics)
  }
}

// ---------------- Kernel 3: combine ----------------------------------------
__global__ void bert_loss_finalize(const float* __restrict__ ws,
                                   float* __restrict__ out, int n) {
  float mrsum = 0.f;
  for (int i = 0; i < MR_BLOCKS; ++i) mrsum += ws[4 + i];
  const float nn   = (float)n;
  const float mse  = ws[0] / nn;
  const float mr   = mrsum / (nn * nn);
  const float den  = fmaxf(sqrtf(ws[2]) * sqrtf(ws[3]), COS_EPS);
  const float sim  = 1.0f - ws[1] / den;
  out[0] = mse + mr + sim;       // ALPHA=BETA=GAMMA=1
}

// ---------------------------------------------------------------------------
extern "C" void kernel_launch(void* const* d_in, const int* in_sizes, int n_in,
                              void* d_out, int out_size, void* d_ws, size_t ws_size,
                              hipStream_t stream) {
  const float* p = (const float*)d_in[0];   // predictions  (float32, B)
  const float* g = (const float*)d_in[1];   // correct_output (float32, B)
  float* out = (float*)d_out;               // loss, shape (1,)
  float* ws  = (float*)d_ws;                // [0..3]=reductions, [4..4+MR_BLOCKS)=partials
  const int n = in_sizes[0];

  bert_loss_reduce<<<1, THREADS, 0, stream>>>(p, g, n, ws);
  bert_loss_mr_pairwise<<<MR_BLOCKS, THREADS, 0, stream>>>(p, n, ws);
  bert_loss_finalize<<<1, 1, 0, stream>>>(ws, out, n);
}